// SSM_9122510536957
// MI455X (gfx1250) — compile-verified
//
#include <hip/hip_runtime.h>
#include <hip/hip_bf16.h>

typedef __attribute__((ext_vector_type(2))) float v2f;
typedef __attribute__((ext_vector_type(8))) float v8f;

// Problem constants (from reference setup_inputs)
#define BATCH 16
#define CIN   64
#define LLEN  1024
#define HDIM  256
#define NDIM  64

// Workspace layout (float offsets)
#define OFF_A   0        // a[64]   : diag of A
#define OFF_G   64       // G[64]   : sum_l a^l
#define OFF_E0  128      // e0[64]  : Bm @ b_up
#define OFF_E   256      // E[64*64]: Bm @ W_up
#define OFF_T   4352     // t[16*64]
#define OFF_W   8192     // w[64*1024] : w[n,l] = a_n^l
#define OFF_S   73728    // S[16*64*64]: S[b][n][c]

// ---------------------------------------------------------------------------
// Kernel 1: a[n] = A[n,n]; w[n,l] = a^l (repeated fp32 multiply, matching the
// reference scan's rounding); G[n] = sum_l a^l.
// ---------------------------------------------------------------------------
__global__ __launch_bounds__(64)
void k_pow(const float* __restrict__ A, float* __restrict__ ws) {
    int n = threadIdx.x;                    // 0..63
    float a = A[n * NDIM + n];
    float p = 1.0f, g = 0.0f;
    float* wrow = ws + OFF_W + n * LLEN;
    for (int l = 0; l < LLEN; ++l) {
        wrow[l] = p;
        g += p;
        p *= a;
    }
    ws[OFF_A + n] = a;
    ws[OFF_G + n] = g;
}

// ---------------------------------------------------------------------------
// Kernel 2: E[n,c] = sum_h Bm[n,h] * W_up[h,c];  e0[n] = sum_h Bm[n,h]*b_up[h]
// grid = 64 (n), block = 64 (c)
// ---------------------------------------------------------------------------
__global__ __launch_bounds__(64)
void k_fold(const float* __restrict__ Bm, const float* __restrict__ W_up,
            const float* __restrict__ b_up, float* __restrict__ ws) {
    int n = blockIdx.x;
    int c = threadIdx.x;
    const float* brow = Bm + n * HDIM;
    float acc = 0.0f;
    #pragma unroll 8
    for (int h = 0; h < HDIM; ++h)
        acc += brow[h] * W_up[h * CIN + c];
    ws[OFF_E + n * CIN + c] = acc;
    if (c == 0) {
        float e0 = 0.0f;
        for (int h = 0; h < HDIM; ++h) e0 += brow[h] * b_up[h];
        ws[OFF_E0 + n] = e0;
    }
}

// ---------------------------------------------------------------------------
// Kernel 3 (WMMA): S[b][n][c] = sum_l u[b][c][l] * w[n][l]
// One block per batch; 16 waves; wave w -> 16x16 output tile (c0 = (w/4)*16,
// n0 = (w%4)*16). K-loop over L in steps of 4 using V_WMMA_F32_16X16X4_F32.
//
// 16x4 f32 A layout (ISA 7.12.2): lane t holds A[m = t&15][k = (t>>4)*2 + {0,1}]
// in {v0,v1} -> one aligned float2 load. B (4x16) mirrors it: lane t holds
// B[k = (t>>4)*2 + {0,1}][n = t&15] = w[n0 + (t&15)][l + k] -> same pattern.
// D layout: vgpr r, lanes 0-15 -> (M=r, N=lane); lanes 16-31 -> (M=r+8, N=lane-16).
// ---------------------------------------------------------------------------
__global__ __launch_bounds__(512)
void k_gemm_s(const float* __restrict__ u, float* __restrict__ ws) {
    const int b    = blockIdx.x;
    const int wave = threadIdx.x >> 5;
    const int lane = threadIdx.x & 31;
    const int c0   = (wave >> 2) * 16;
    const int n0   = (wave & 3) * 16;
    const int m    = lane & 15;
    const int koff = (lane >> 4) * 2;

    const float* ub = u + b * CIN * LLEN;
    const v2f* pa = reinterpret_cast<const v2f*>(ub + (c0 + m) * LLEN + koff);
    const v2f* pb = reinterpret_cast<const v2f*>(ws + OFF_W + (n0 + m) * LLEN + koff);

    v8f acc = {};
    #pragma unroll 8
    for (int kk = 0; kk < LLEN; kk += 4) {
        v2f av = pa[kk >> 1];
        v2f bv = pb[kk >> 1];
        acc = __builtin_amdgcn_wmma_f32_16x16x4_f32(
            /*neg_a=*/false, av, /*neg_b=*/false, bv,
            /*c_mod=*/(short)0, acc, /*reuse_a=*/false, /*reuse_b=*/false);
    }

    // Scatter D: S[b][n0+nd][c0+md]
    float* S = ws + OFF_S + b * NDIM * CIN;
    const int nd = lane & 15;
    const int mbase = (lane >> 4) * 8;
    #pragma unroll
    for (int r = 0; r < 8; ++r) {
        int md = mbase + r;
        S[(n0 + nd) * CIN + (c0 + md)] = acc[r];
    }
}

// ---------------------------------------------------------------------------
// Kernel 4: t[b,n] = e0[n]*G[n] + sum_c E[n,c]*S[b][n][c]
// grid = 4, block = 256  (idx = b*64 + n)
// ---------------------------------------------------------------------------
__global__ __launch_bounds__(256)
void k_reduce_t(float* __restrict__ ws) {
    int idx = blockIdx.x * 256 + threadIdx.x;   // 0..1023
    int b = idx >> 6;
    int n = idx & 63;
    const float* Erow = ws + OFF_E + n * CIN;
    const float* Srow = ws + OFF_S + (b * NDIM + n) * CIN;
    float acc = ws[OFF_E0 + n] * ws[OFF_G + n];
    #pragma unroll 8
    for (int c = 0; c < CIN; ++c)
        acc += Erow[c] * Srow[c];
    ws[OFF_T + idx] = acc;
}

// ---------------------------------------------------------------------------
// Kernel 5: y[b,i] = sum_n Cm[i,n]*t[b,n]; out[b,j] = b_down[j] + sum_i W_down[j,i]*y[b,i]
// grid = 16 (b), block = 256 (i)
// ---------------------------------------------------------------------------
__global__ __launch_bounds__(256)
void k_head(const float* __restrict__ Cm, const float* __restrict__ W_down,
            const float* __restrict__ b_down, const float* __restrict__ ws,
            float* __restrict__ out) {
    __shared__ float ysh[HDIM];
    int b = blockIdx.x;
    int i = threadIdx.x;
    const float* trow = ws + OFF_T + b * NDIM;
    const float* crow = Cm + i * NDIM;
    float y = 0.0f;
    #pragma unroll 8
    for (int n = 0; n < NDIM; ++n)
        y += crow[n] * trow[n];
    ysh[i] = y;
    __syncthreads();
    if (i < 10) {
        const float* wrow = W_down + i * HDIM;
        float o = b_down[i];
        #pragma unroll 8
        for (int h = 0; h < HDIM; ++h)
            o += wrow[h] * ysh[h];
        out[b * 10 + i] = o;
    }
}

// ---------------------------------------------------------------------------
extern "C" void kernel_launch(void* const* d_in, const int* in_sizes, int n_in,
                              void* d_out, int out_size, void* d_ws, size_t ws_size,
                              hipStream_t stream) {
    const float* u      = (const float*)d_in[0];
    const float* W_up   = (const float*)d_in[1];
    const float* b_up   = (const float*)d_in[2];
    const float* A      = (const float*)d_in[3];
    const float* Bm     = (const float*)d_in[4];
    const float* Cm     = (const float*)d_in[5];
    const float* W_down = (const float*)d_in[6];
    const float* b_down = (const float*)d_in[7];
    float* out = (float*)d_out;
    float* ws  = (float*)d_ws;

    k_pow<<<1, 64, 0, stream>>>(A, ws);
    k_fold<<<64, 64, 0, stream>>>(Bm, W_up, b_up, ws);
    k_gemm_s<<<BATCH, 512, 0, stream>>>(u, ws);
    k_reduce_t<<<4, 256, 0, stream>>>(ws);
    k_head<<<BATCH, 256, 0, stream>>>(Cm, W_down, b_down, ws, out);
}